// SpectralConv1d_80685255622855
// MI455X (gfx1250) — compile-verified
//
#include <hip/hip_runtime.h>
#include <cstddef>
#include <cstdint>

// ---------------------------------------------------------------------------
// SpectralConv1d pipeline, algebraically folded:
//   XH = X * F^T            (1024 x 16384 x 64, f32 WMMA, split-K, TDM staging)
//   OM = compl_mul1d(XH,W)  (tiny VALU)
//   SM = wrap-gauss over (b,o) of OM, * 1/N   (axis-2 gauss folded into G basis)
//   Y  = SM * G^T, masked   (1024 x 64 x 16384, f32 WMMA, dead tiles skipped)
// ---------------------------------------------------------------------------

#define L_LEN   16384
#define N_SEG   8
#define SEG_LEN 2048
#define CHUNK   128
#define SP      132          // padded LDS row stride (floats): 4*row distinct mod 64

typedef __attribute__((ext_vector_type(2))) float v2f;
typedef __attribute__((ext_vector_type(8))) float v8f;
typedef unsigned int u32;
typedef __attribute__((ext_vector_type(4))) u32 u32x4;
typedef __attribute__((ext_vector_type(4))) int i32x4;
typedef __attribute__((ext_vector_type(8))) int i32x8;

#define HAVE_TDM __has_builtin(__builtin_amdgcn_tensor_load_to_lds)

__constant__ const float TWO_PI = 6.28318530717958647692f;

#if HAVE_TDM
// Issue one TDM DMA: 16 rows x 128 f32, global row stride 16384 f32 -> LDS at
// byte offset lds_off, with HW padding of 4 DWORDs after every 128 DWORDs so
// the LDS row stride is SP=132 floats (bank-conflict-free for the A-fragment
// reads). D# bitfields per CDNA5 ISA 8.3/8.4.
__device__ __forceinline__ void tdm_load_tile(const float* gsrc, u32 lds_off) {
  uint64_t ga = (uint64_t)(uintptr_t)gsrc;
  u32x4 g0;
  g0.x = 1u;                                              // count=1 user descriptor
  g0.y = lds_off;                                         // lds_addr (bytes)
  g0.z = (u32)ga;                                         // global_addr[31:0]
  g0.w = ((u32)(ga >> 32) & 0x01FFFFFFu) | (2u << 30);    // addr[56:32] | type=2
  i32x8 g1;
  g1[0] = (int)((2u << 16) |        // data_size = 4 bytes
                (1u << 20) |        // pad_enable
                (6u << 22) |        // pad_interval: 128 DWORDs
                (3u << 25));        // pad_amount: 4 DWORDs
  g1[1] = (int)(16384u << 16);      // tensor_dim0 = 16384       (bits [79:48])
  g1[2] = (int)(1024u << 16);       // tensor_dim1 = 1024        (bits [111:80])
  g1[3] = (int)(128u << 16);        // tile_dim0 = 128           (bits [127:112])
  g1[4] = 16;                       // tile_dim1 = 16            (bits [143:128])
  g1[5] = 16384;                    // tensor_dim0_stride = 16384 (bits [207:160])
  g1[6] = (int)(16384u << 16);      // tensor_dim1_stride = 16384 (bits [255:208])
  g1[7] = 0;
  i32x4 z4 = {0, 0, 0, 0};
#if defined(__clang_major__) && (__clang_major__ >= 23)
  i32x8 z8 = {0, 0, 0, 0, 0, 0, 0, 0};
  __builtin_amdgcn_tensor_load_to_lds(g0, g1, z4, z4, z8, 0);
#else
  __builtin_amdgcn_tensor_load_to_lds(g0, g1, z4, z4, 0);
#endif
}
#endif  // HAVE_TDM

// ---- basis builder: Ft[n*64+k] forward, G[n*64+k] inverse*ghat ----
__global__ void k_basis(float* __restrict__ Ft, float* __restrict__ G) {
  int idx = blockIdx.x * blockDim.x + threadIdx.x;   // 16384*64 threads
  int n = idx >> 6;
  int k = idx & 63;
  const float inv = 1.0f / 16384.0f;
  float c  = cosf(TWO_PI * (float)((k * n) & 16383) * inv);        // cos(2pi kn/N)
  float sF = sinf(TWO_PI * (float)((k * (n + 1)) & 16383) * inv);  // forward sin
  float sG = sinf(TWO_PI * (float)((n * (k + 1)) & 16383) * inv);  // inverse sin
  // normalized gaussian taps (sigma=1, radius=4)
  float e1 = expf(-0.5f), e2 = expf(-2.0f), e3 = expf(-4.5f), e4 = expf(-8.0f);
  float z  = 1.0f + 2.0f * (e1 + e2 + e3 + e4);
  float w0 = 1.0f / z; e1 /= z; e2 /= z; e3 /= z; e4 /= z;
  // ghat(k), ghat(k+1): eigenvalues of the circular gaussian along axis 2
  float a1 = TWO_PI * (float)k * inv;
  float gk  = w0 + 2.0f * (e1 * cosf(a1) + e2 * cosf(2.0f * a1) +
                           e3 * cosf(3.0f * a1) + e4 * cosf(4.0f * a1));
  float a2 = TWO_PI * (float)(k + 1) * inv;
  float gk1 = w0 + 2.0f * (e1 * cosf(a2) + e2 * cosf(2.0f * a2) +
                           e3 * cosf(3.0f * a2) + e4 * cosf(4.0f * a2));
  Ft[(size_t)n * 64 + k] = c - sF;
  G[(size_t)n * 64 + k]  = gk * c - gk1 * sG;
}

// ---- forward GEMM: XHp[seg][r][k] = sum_{n in seg} X[r][n] * Ft[n][k] ----
__global__ void k_fwd(const float* __restrict__ X, const float* __restrict__ Ft,
                      float* __restrict__ XHp) {
  __shared__ float xs[2][16 * SP];
  const int blk  = blockIdx.x;          // 512 blocks
  const int seg  = blk & (N_SEG - 1);
  const int r0   = (blk >> 3) * 16;
  const int tid  = threadIdx.x;         // 128 threads = 4 waves (wave32)
  const int wave = tid >> 5;
  const int lane = tid & 31;
  const int l15  = lane & 15;
  const int kh   = lane >> 4;           // K half-select per WMMA A/B layout
  const int k0   = wave * 16;
  const int nbase = seg * SEG_LEN;
  const int NIT   = SEG_LEN / CHUNK;    // 16
  v8f acc = {};
#if HAVE_TDM
  // TDM writes xs behind the compiler's back. Plant a dynamically-dead store
  // (opaque-zero guard) so alias analysis must treat xs as written, keeping
  // the addrspace(3) ds_loads below alive — without genericizing the pointer.
  {
    u32 zero = 0;
    asm volatile("" : "+s"(zero));
    if (zero) xs[0][tid] = 0.0f;
  }
  if (wave == 0)
    tdm_load_tile(X + (size_t)r0 * L_LEN + nbase, 0u);
#endif
  for (int it = 0; it < NIT; ++it) {
    const int n0  = nbase + it * CHUNK;
    const int buf = it & 1;
#if HAVE_TDM
    if (wave == 0) {
      if (it + 1 < NIT) {
        // double-buffer: kick next tile, wait for current (in-order TENSORcnt)
        tdm_load_tile(X + (size_t)r0 * L_LEN + (n0 + CHUNK),
                      (u32)(((it + 1) & 1) * 16 * SP * 4));
        __builtin_amdgcn_s_wait_tensorcnt((short)1);
      } else {
        __builtin_amdgcn_s_wait_tensorcnt((short)0);
      }
    }
    __syncthreads();
#else
    // fallback: manual coalesced staging
#pragma unroll
    for (int q = 0; q < 4; ++q) {
      int f   = (q * 128 + tid) << 2;
      int row = f >> 7;
      int col = f & 127;
      float4 v = *(const float4*)(X + (size_t)(r0 + row) * L_LEN + (n0 + col));
      float* p = &xs[buf][row * SP + col];
      p[0] = v.x; p[1] = v.y; p[2] = v.z; p[3] = v.w;
    }
    __syncthreads();
#endif
    const float* bbase = Ft + (size_t)n0 * 64 + k0 + l15;
    const float* ap0   = &xs[buf][l15 * SP + 2 * kh];
#pragma unroll 8
    for (int nn = 0; nn < CHUNK; nn += 4) {
      v2f a, b;
      a.x = ap0[nn];
      a.y = ap0[nn + 1];
      const float* bp = bbase + (size_t)(nn + 2 * kh) * 64;
      b.x = bp[0];
      b.y = bp[64];
      acc = __builtin_amdgcn_wmma_f32_16x16x4_f32(false, a, false, b, (short)0,
                                                  acc, false, false);
    }
    __syncthreads();
  }
  // D layout: VGPR j -> row j + 8*kh, col = lane&15
  float* dst = XHp + ((size_t)seg * 1024 + r0) * 64 + k0 + l15;
#pragma unroll
  for (int j = 0; j < 8; ++j)
    dst[(size_t)(j + 8 * kh) * 64] = acc[j];
}

// ---- split-K reduce: XH[rk] = sum_seg XHp[seg][rk] ----
__global__ void k_reduce(const float* __restrict__ XHp, float* __restrict__ XH) {
  int idx = blockIdx.x * blockDim.x + threadIdx.x;   // 65536 threads
  float s = 0.0f;
#pragma unroll
  for (int sg = 0; sg < N_SEG; ++sg) s += XHp[(size_t)sg * 65536 + idx];
  XH[idx] = s;
}

// ---- compl_mul1d: OM[b,o,m] = 0.5*(x.w - xn.wn + x.wn + xn.w) summed over i ----
__global__ void k_mix(const float* __restrict__ XH, const float* __restrict__ W,
                      float* __restrict__ OM) {
  int idx = blockIdx.x * blockDim.x + threadIdx.x;   // 65536 threads
  int b  = idx >> 12;
  int o  = (idx >> 6) & 63;
  int m  = idx & 63;
  int mn = (64 - m) & 63;
  const float* xr = XH + (size_t)b * 64 * 64;
  float acc = 0.0f;
  for (int i = 0; i < 64; ++i) {
    float xv  = xr[i * 64 + m];
    float xnv = xr[i * 64 + mn];
    float wv  = W[((size_t)i * 64 + o) * 64 + m];
    float wnv = W[((size_t)i * 64 + o) * 64 + mn];
    acc += 0.5f * (xv * wv - xnv * wnv + xv * wnv + xnv * wv);
  }
  OM[idx] = acc;
}

// ---- wrap gaussian over (batch, channel) axes + 1/N ----
__global__ void k_smooth(const float* __restrict__ OM, float* __restrict__ SM) {
  int idx = blockIdx.x * blockDim.x + threadIdx.x;   // 65536 threads
  int b = idx >> 12;
  int o = (idx >> 6) & 63;
  int m = idx & 63;
  float e1 = expf(-0.5f), e2 = expf(-2.0f), e3 = expf(-4.5f), e4 = expf(-8.0f);
  float z  = 1.0f + 2.0f * (e1 + e2 + e3 + e4);
  float w4[5] = {1.0f / z, e1 / z, e2 / z, e3 / z, e4 / z};
  float acc = 0.0f;
  for (int p = -4; p <= 4; ++p) {
    int bb   = (b + p + 16) & 15;
    float wp = w4[p < 0 ? -p : p];
    for (int q = -4; q <= 4; ++q) {
      int oo = (o + q + 64) & 63;
      acc += wp * w4[q < 0 ? -q : q] * OM[((size_t)bb * 64 + oo) * 64 + m];
    }
  }
  SM[idx] = acc * (1.0f / 16384.0f);
}

// ---- synthesis GEMM + low-pass mask: Y[r,n] = mask(n) * sum_k SM[r,k]*G[n,k] ----
__global__ void k_syn(const float* __restrict__ SM, const float* __restrict__ G,
                      float* __restrict__ Y) {
  const int blk  = blockIdx.x;          // 16384 blocks
  const int rt   = blk >> 8;            // 0..63 row tile
  const int ntg  = blk & 255;
  const int tid  = threadIdx.x;         // 128 threads = 4 waves
  const int wave = tid >> 5;
  const int lane = tid & 31;
  const int l15  = lane & 15;
  const int kh   = lane >> 4;
  const int n0   = (ntg * 4 + wave) * 16;
  const int r0   = rt * 16;
  const int n    = n0 + l15;
  float* ybase = Y + (size_t)r0 * L_LEN + n;
  // wave-uniform dead-tile test (EXEC stays all-ones inside each wave)
  const bool live = (n0 <= 4096) || (n0 + 15 >= 12288);
  if (!live) {
#pragma unroll
    for (int j = 0; j < 8; ++j)
      ybase[(size_t)(j + 8 * kh) * L_LEN] = 0.0f;
    return;
  }
  v8f acc = {};
  const float* ap = SM + (size_t)(r0 + l15) * 64 + 2 * kh;
  const float* bp = G + (size_t)n * 64 + 2 * kh;
#pragma unroll
  for (int kk = 0; kk < 64; kk += 4) {
    v2f a = *(const v2f*)(ap + kk);
    v2f b = *(const v2f*)(bp + kk);
    acc = __builtin_amdgcn_wmma_f32_16x16x4_f32(false, a, false, b, (short)0,
                                                acc, false, false);
  }
  const float msk = (n <= 4096 || n >= 12288) ? 1.0f : 0.0f;
#pragma unroll
  for (int j = 0; j < 8; ++j)
    ybase[(size_t)(j + 8 * kh) * L_LEN] = acc[j] * msk;
}

extern "C" void kernel_launch(void* const* d_in, const int* in_sizes, int n_in,
                              void* d_out, int out_size, void* d_ws, size_t ws_size,
                              hipStream_t stream) {
  (void)in_sizes; (void)n_in; (void)out_size; (void)ws_size;
  const float* X = (const float*)d_in[0];   // [16, 64, 16384]
  const float* W = (const float*)d_in[1];   // [64, 64, 64]
  float* Y  = (float*)d_out;                // [16, 64, 16384]
  float* ws = (float*)d_ws;
  float* Ft  = ws;                 // 1,048,576 f32 (4 MB) forward basis [n][k]
  float* G   = ws + 1048576;       // 1,048,576 f32 (4 MB) inverse basis [n][k]
  float* XHp = ws + 2097152;       //   524,288 f32 split-K partials [seg][r][k]
  float* XH  = ws + 2621440;       //    65,536 f32 modes [r][k]
  float* OM  = ws + 2686976;       //    65,536 f32 mixed modes
  float* SM  = ws + 2752512;       //    65,536 f32 smoothed modes

  k_basis <<<4096, 256, 0, stream>>>(Ft, G);
  k_fwd   <<<512, 128, 0, stream>>>(X, Ft, XHp);
  k_reduce<<<256, 256, 0, stream>>>(XHp, XH);
  k_mix   <<<256, 256, 0, stream>>>(XH, W, OM);
  k_smooth<<<256, 256, 0, stream>>>(OM, SM);
  k_syn   <<<16384, 128, 0, stream>>>(SM, G, Y);
}